// OrthonormalTransform_30537217475132
// MI455X (gfx1250) — compile-verified
//
#include <hip/hip_runtime.h>
#include <math.h>

typedef __attribute__((ext_vector_type(2))) float v2f;
typedef __attribute__((ext_vector_type(8))) float v8f;

#define NPT 64

// ---------------------------------------------------------------------------
// Kernel 1: build R = diag(mus) * (G_{2015} ... G_0).
// Key insight: each Givens step only mixes rows (t,b) within the same column,
// so each lane owns one column of R in LDS and runs the whole 2016-step chain
// privately -> no barriers, single 64-thread block, 2 waves.
// ---------------------------------------------------------------------------
__global__ __launch_bounds__(NPT) void build_R_kernel(
    const float* __restrict__ angles, const float* __restrict__ mus,
    float* __restrict__ R /* 64x64 row-major, in d_ws */) {
  __shared__ float Rs[NPT][NPT];
  const int col = threadIdx.x;  // 0..63, exclusive column owner
  #pragma unroll 4
  for (int r = 0; r < NPT; ++r) Rs[r][col] = (r == col) ? 1.0f : 0.0f;

  int idx = 0;
  for (int t = 0; t < NPT - 1; ++t) {
    for (int b = t + 1; b < NPT; ++b, ++idx) {
      float ang = angles[idx];
      float c = cosf(ang);
      float s = sinf(ang);
      float rt = Rs[t][col];
      float rb = Rs[b][col];
      Rs[t][col] = c * rt - s * rb;
      Rs[b][col] = s * rt + c * rb;
    }
  }
  #pragma unroll 4
  for (int r = 0; r < NPT; ++r) R[r * NPT + col] = mus[r] * Rs[r][col];
}

// ---------------------------------------------------------------------------
// Kernel 2: Y = R @ X with V_WMMA_F32_16X16X4_F32.
// Per wave: 64-row x 16-col slab = 4 C tiles; K=64 in 16 chunks of 4.
// A (R) lives in registers (4x16 v2f), loaded once per wave, reused across
// the grid-stride slab loop. B built with the ISA 4x16 f32 layout:
//   VGPR0: lanes 0-15 K=0, lanes 16-31 K=2 ; VGPR1: K=1 / K=3.
// ---------------------------------------------------------------------------
__global__ __launch_bounds__(256) void givens_gemm_kernel(
    const float* __restrict__ R, const float* __restrict__ X,
    float* __restrict__ Y, int M, int nSlabs, int nWaves) {
  const int lane = threadIdx.x & 31;
  const int wave = blockIdx.x * (blockDim.x >> 5) + (threadIdx.x >> 5);
  const int mrow = lane & 15;  // row within 16-row tile / column within slab
  const int h    = lane >> 4;  // K half-select (0 -> K{0,1}, 1 -> K{2,3})

  // A tiles of R, register resident: a[m][kc] covers rows m*16..+15, K 4*kc..+3
  v2f a[4][16];
  #pragma unroll
  for (int m = 0; m < 4; ++m) {
    const float* ra = R + (size_t)(m * 16 + mrow) * NPT + 2 * h;
    #pragma unroll
    for (int kc = 0; kc < 16; ++kc)
      a[m][kc] = *reinterpret_cast<const v2f*>(ra + kc * 4);
  }

  for (int slab = wave; slab < nSlabs; slab += nWaves) {  // wave-uniform loop
    const size_t n0 = (size_t)slab * 16;
    v8f c0 = {}, c1 = {}, c2 = {}, c3 = {};

    const float* xb = X + (size_t)(2 * h) * M + n0 + mrow;
    #pragma unroll
    for (int kc = 0; kc < 16; ++kc) {
      const float* xp = xb + (size_t)(kc * 4) * M;
      v2f b;
      b.x = xp[0];           // K = 4*kc + 2h
      b.y = xp[(size_t)M];   // K = 4*kc + 2h + 1
      c0 = __builtin_amdgcn_wmma_f32_16x16x4_f32(false, a[0][kc], false, b,
                                                 (short)0, c0, false, false);
      c1 = __builtin_amdgcn_wmma_f32_16x16x4_f32(false, a[1][kc], false, b,
                                                 (short)0, c1, false, false);
      c2 = __builtin_amdgcn_wmma_f32_16x16x4_f32(false, a[2][kc], false, b,
                                                 (short)0, c2, false, false);
      c3 = __builtin_amdgcn_wmma_f32_16x16x4_f32(false, a[3][kc], false, b,
                                                 (short)0, c3, false, false);
    }

    // C/D layout: VGPR r -> row m*16 + r + 8*h, col n0 + (lane&15)
    float* yp = Y + (size_t)(8 * h) * M + n0 + mrow;
    #pragma unroll
    for (int r = 0; r < 8; ++r) {
      yp[(size_t)(r)      * M] = c0[r];
      yp[(size_t)(16 + r) * M] = c1[r];
      yp[(size_t)(32 + r) * M] = c2[r];
      yp[(size_t)(48 + r) * M] = c3[r];
    }
  }
}

// Scalar cleanup for M % 16 columns (M = 300000 -> unused, kept for safety).
__global__ __launch_bounds__(NPT) void gemm_tail_kernel(
    const float* __restrict__ R, const float* __restrict__ X,
    float* __restrict__ Y, int M, int tailStart) {
  const int col = tailStart + blockIdx.x;
  const int row = threadIdx.x;
  if (col >= M) return;
  float acc = 0.0f;
  #pragma unroll 8
  for (int k = 0; k < NPT; ++k)
    acc = fmaf(R[row * NPT + k], X[(size_t)k * M + col], acc);
  Y[(size_t)row * M + col] = acc;
}

extern "C" void kernel_launch(void* const* d_in, const int* in_sizes, int n_in,
                              void* d_out, int out_size, void* d_ws,
                              size_t ws_size, hipStream_t stream) {
  const float* X      = (const float*)d_in[0];  // (64, M) fp32
  const float* angles = (const float*)d_in[1];  // (2016,) fp32
  const float* mus    = (const float*)d_in[2];  // (64,)   fp32
  float* Y = (float*)d_out;                     // (64, M) fp32
  float* R = (float*)d_ws;                      // 64*64 fp32 scratch (16 KB)

  const int M = in_sizes[0] / NPT;

  build_R_kernel<<<1, NPT, 0, stream>>>(angles, mus, R);

  const int nSlabs = M / 16;
  if (nSlabs > 0) {
    int blocks = (nSlabs + 15) / 16;           // 8 waves/block, ~2 slabs/wave
    if (blocks < 1) blocks = 1;
    const int nWaves = blocks * 8;
    givens_gemm_kernel<<<blocks, 256, 0, stream>>>(R, X, Y, M, nSlabs, nWaves);
  }

  const int tail = M % 16;
  if (tail > 0)
    gemm_tail_kernel<<<tail, NPT, 0, stream>>>(R, X, Y, M, (M / 16) * 16);
}